// LandmarkGCN_78297253806903
// MI455X (gfx1250) — compile-verified
//
#include <hip/hip_runtime.h>
#include <hip/hip_bf16.h>

// ---- problem constants (match reference) ----
#define N_NODES   200000
#define N_EDGES   3200000
#define HIDDEN    64
#define N_GRAPHS  4096
#define IN_CH     2
#define N_CLASSES 7

typedef __attribute__((ext_vector_type(2))) float v2f;
typedef __attribute__((ext_vector_type(8))) float v8f;

static __host__ int cdiv(long long a, long long b) { return (int)((a + b - 1) / b); }

// -------------------- small utility kernels --------------------

__global__ void fill_f32(float* __restrict__ p, float v, int n) {
  int t = blockIdx.x * blockDim.x + threadIdx.x;
  if (t < n) p[t] = v;
}

// deg starts at 1.0 (self loops); add 1 per incoming edge at dst
__global__ void degree_accum(const long long* __restrict__ dst, float* __restrict__ deg, int n_edges) {
  int e = blockIdx.x * blockDim.x + threadIdx.x;
  if (e < n_edges) atomicAdd(&deg[(int)dst[e]], 1.0f);
}

// deg -> deg^-1/2 in place (deg >= 1 always due to self loops)
__global__ void deg_to_dinv(float* __restrict__ deg, int n) {
  int t = blockIdx.x * blockDim.x + threadIdx.x;
  if (t < n) deg[t] = rsqrtf(deg[t]);
}

// layer 1: xw[n, j] = x[n,0]*W1[0,j] + x[n,1]*W1[1,j]   (K=2, not worth WMMA)
__global__ void layer1_xw(const float* __restrict__ x, const float* __restrict__ W1,
                          float* __restrict__ xw, int n_nodes) {
  int t = blockIdx.x * blockDim.x + threadIdx.x;
  if (t >= n_nodes * HIDDEN) return;
  int n = t >> 6, j = t & 63;
  xw[t] = x[n * 2 + 0] * W1[j] + x[n * 2 + 1] * W1[HIDDEN + j];
}

// edge scatter: 64 consecutive threads handle one edge's 64 channels
// acc[dst, j] += dinv[src]*dinv[dst] * xw[src, j]
__global__ void edge_scatter(const long long* __restrict__ src, const long long* __restrict__ dst,
                             const float* __restrict__ dinv, const float* __restrict__ xw,
                             float* __restrict__ acc, int n_edges) {
  int t = blockIdx.x * blockDim.x + threadIdx.x;
  if (t >= n_edges * HIDDEN) return;
  int e = t >> 6, j = t & 63;
  int s = (int)src[e];
  int d = (int)dst[e];
  float w = dinv[s] * dinv[d];
  atomicAdd(&acc[(size_t)d * HIDDEN + j], w * xw[(size_t)s * HIDDEN + j]);
}

// self loop + bias + relu (in place on accumulator):
// acc[n,j] = relu(acc[n,j] + dinv[n]^2 * xw[n,j] + b[j])
__global__ void selfloop_bias_relu(float* __restrict__ acc, const float* __restrict__ xw,
                                   const float* __restrict__ dinv, const float* __restrict__ bias,
                                   int n_nodes) {
  int t = blockIdx.x * blockDim.x + threadIdx.x;
  if (t >= n_nodes * HIDDEN) return;
  int n = t >> 6, j = t & 63;
  float di = dinv[n];
  float v = acc[t] + di * di * xw[t] + bias[j];
  acc[t] = fmaxf(v, 0.0f);
}

// -------------------- WMMA f32 GEMM: out[M,64] = A[M,64] @ W[64,64] --------------------
// One wave computes one 16x16 output tile via 16 steps of V_WMMA_F32_16X16X4_F32.
// A-frag layout (32-bit A 16x4): lane L<16 holds M=L, {K0,K1}; lane L>=16 holds M=L-16, {K2,K3}.
// B-frag layout (4x16):          lane L<16 holds N=L, rows {K0,K1}; high lanes rows {K2,K3}.
// C/D layout: c[v] = C[m0 + v + 8*half, n0 + (lane&15)].
__global__ void gemm64_wmma(const float* __restrict__ A, const float* __restrict__ W,
                            const float* __restrict__ bias, float* __restrict__ out,
                            int M, int do_relu) {
  int wave = (blockIdx.x * blockDim.x + threadIdx.x) >> 5;
  int lane = threadIdx.x & 31;
  int n_mtiles = M >> 4;            // M is a multiple of 16 here
  int mtile = wave >> 2;            // 4 column tiles of 16 (HIDDEN=64)
  int ntile = wave & 3;
  if (mtile >= n_mtiles) return;
  int m0 = mtile << 4;
  int n0 = ntile << 4;
  int half = lane >> 4;             // 0: K pair {0,1}; 1: K pair {2,3}
  int l = lane & 15;

  const float* arow = A + (size_t)(m0 + l) * HIDDEN + 2 * half; // contiguous float2 per step
  const float* bcol = W + (size_t)(2 * half) * HIDDEN + n0 + l; // column-strided

  v8f c = {};
#pragma unroll
  for (int k = 0; k < HIDDEN; k += 4) {
    v2f a, b;
    a.x = arow[k + 0];
    a.y = arow[k + 1];
    b.x = bcol[(size_t)(k + 0) * HIDDEN];
    b.y = bcol[(size_t)(k + 1) * HIDDEN];
    // 8 args: (neg_a, A, neg_b, B, c_mod, C, reuse_a, reuse_b)
    c = __builtin_amdgcn_wmma_f32_16x16x4_f32(false, a, false, b, (short)0, c, false, false);
  }

  float bv = bias ? bias[n0 + l] : 0.0f;
  float* orow = out + (size_t)(m0 + 8 * half) * HIDDEN + n0 + l;
#pragma unroll
  for (int v = 0; v < 8; ++v) {
    float val = c[v] + bv;
    if (do_relu) val = fmaxf(val, 0.0f);
    orow[(size_t)v * HIDDEN] = val;
  }
}

// -------------------- pooling --------------------

__global__ void graph_count(const long long* __restrict__ batch, float* __restrict__ counts, int n_nodes) {
  int n = blockIdx.x * blockDim.x + threadIdx.x;
  if (n < n_nodes) atomicAdd(&counts[(int)batch[n]], 1.0f);
}

__global__ void pool_sum(const long long* __restrict__ batch, const float* __restrict__ h,
                         float* __restrict__ pooled, int n_nodes) {
  int t = blockIdx.x * blockDim.x + threadIdx.x;
  if (t >= n_nodes * HIDDEN) return;
  int n = t >> 6, j = t & 63;
  int g = (int)batch[n];
  atomicAdd(&pooled[(size_t)g * HIDDEN + j], h[t]);
}

__global__ void pool_div(float* __restrict__ pooled, const float* __restrict__ counts, int n_graphs) {
  int t = blockIdx.x * blockDim.x + threadIdx.x;
  if (t >= n_graphs * HIDDEN) return;
  int g = t >> 6;
  pooled[t] /= fmaxf(counts[g], 1.0f);
}

// -------------------- classifier head (64 -> 7) --------------------

__global__ void head_out(const float* __restrict__ hid, const float* __restrict__ Wf2,
                         const float* __restrict__ bf2, float* __restrict__ out, int n_graphs) {
  int t = blockIdx.x * blockDim.x + threadIdx.x;
  if (t >= n_graphs * N_CLASSES) return;
  int g = t / N_CLASSES, c = t % N_CLASSES;
  float s = bf2[c];
  const float* hrow = hid + (size_t)g * HIDDEN;
#pragma unroll
  for (int j = 0; j < HIDDEN; ++j) s += hrow[j] * Wf2[j * N_CLASSES + c];
  out[t] = s;
}

// -------------------- launch --------------------

extern "C" void kernel_launch(void* const* d_in, const int* in_sizes, int n_in,
                              void* d_out, int out_size, void* d_ws, size_t ws_size,
                              hipStream_t stream) {
  (void)in_sizes; (void)n_in; (void)out_size; (void)ws_size;

  const float*     x     = (const float*)d_in[0];
  const long long* ei    = (const long long*)d_in[1];       // [2, N_EDGES]
  const long long* batch = (const long long*)d_in[2];
  const float* W1  = (const float*)d_in[3];
  const float* b1  = (const float*)d_in[4];
  const float* W2  = (const float*)d_in[5];
  const float* b2  = (const float*)d_in[6];
  const float* W3  = (const float*)d_in[7];
  const float* b3  = (const float*)d_in[8];
  const float* Wf1 = (const float*)d_in[9];
  const float* bf1 = (const float*)d_in[10];
  const float* Wf2 = (const float*)d_in[11];
  const float* bf2 = (const float*)d_in[12];
  float* out = (float*)d_out;

  const long long* esrc = ei;
  const long long* edst = ei + N_EDGES;

  // workspace layout (all offsets 256B-aligned)
  char* ws = (char*)d_ws;
  const size_t NB = (size_t)N_NODES * HIDDEN * sizeof(float);   // 51.2 MB
  float* dinv   = (float*)(ws);                                  // N_NODES floats
  float* bufA   = (float*)(ws + (1u << 20));                     // xw buffer
  float* bufB   = (float*)(ws + (1u << 20) + NB);                // accumulator / h
  float* pooled = (float*)(ws + (1u << 20) + 2 * NB);            // 4096*64
  float* counts = (float*)(ws + (1u << 20) + 2 * NB + (size_t)N_GRAPHS * HIDDEN * sizeof(float));
  float* hid    = (float*)((char*)counts + ((size_t)N_GRAPHS * sizeof(float) + 255) / 256 * 256);

  const int B = 256;
  const int nNH   = N_NODES * HIDDEN;     // 12.8M
  const int nEH   = N_EDGES * HIDDEN;     // 204.8M
  const int gemmM_threads = (N_NODES / 16) * 4 * 32;   // waves*(32 lanes)
  const int gemmG_threads = (N_GRAPHS / 16) * 4 * 32;

  // normalization: deg (init 1 for self loop) -> dinv
  fill_f32<<<cdiv(N_NODES, B), B, 0, stream>>>(dinv, 1.0f, N_NODES);
  degree_accum<<<cdiv(N_EDGES, B), B, 0, stream>>>(edst, dinv, N_EDGES);
  deg_to_dinv<<<cdiv(N_NODES, B), B, 0, stream>>>(dinv, N_NODES);

  // ---- layer 1 ----
  layer1_xw<<<cdiv(nNH, B), B, 0, stream>>>(x, W1, bufA, N_NODES);
  fill_f32<<<cdiv(nNH, B), B, 0, stream>>>(bufB, 0.0f, nNH);
  edge_scatter<<<cdiv(nEH, B), B, 0, stream>>>(esrc, edst, dinv, bufA, bufB, N_EDGES);
  selfloop_bias_relu<<<cdiv(nNH, B), B, 0, stream>>>(bufB, bufA, dinv, b1, N_NODES);

  // ---- layer 2 ----
  gemm64_wmma<<<cdiv(gemmM_threads, B), B, 0, stream>>>(bufB, W2, nullptr, bufA, N_NODES, 0);
  fill_f32<<<cdiv(nNH, B), B, 0, stream>>>(bufB, 0.0f, nNH);
  edge_scatter<<<cdiv(nEH, B), B, 0, stream>>>(esrc, edst, dinv, bufA, bufB, N_EDGES);
  selfloop_bias_relu<<<cdiv(nNH, B), B, 0, stream>>>(bufB, bufA, dinv, b2, N_NODES);

  // ---- layer 3 ----
  gemm64_wmma<<<cdiv(gemmM_threads, B), B, 0, stream>>>(bufB, W3, nullptr, bufA, N_NODES, 0);
  fill_f32<<<cdiv(nNH, B), B, 0, stream>>>(bufB, 0.0f, nNH);
  edge_scatter<<<cdiv(nEH, B), B, 0, stream>>>(esrc, edst, dinv, bufA, bufB, N_EDGES);
  selfloop_bias_relu<<<cdiv(nNH, B), B, 0, stream>>>(bufB, bufA, dinv, b3, N_NODES);

  // ---- mean pool ----
  fill_f32<<<cdiv(N_GRAPHS * HIDDEN, B), B, 0, stream>>>(pooled, 0.0f, N_GRAPHS * HIDDEN);
  fill_f32<<<cdiv(N_GRAPHS, B), B, 0, stream>>>(counts, 0.0f, N_GRAPHS);
  graph_count<<<cdiv(N_NODES, B), B, 0, stream>>>(batch, counts, N_NODES);
  pool_sum<<<cdiv(nNH, B), B, 0, stream>>>(batch, bufB, pooled, N_NODES);
  pool_div<<<cdiv(N_GRAPHS * HIDDEN, B), B, 0, stream>>>(pooled, counts, N_GRAPHS);

  // ---- MLP head ----
  gemm64_wmma<<<cdiv(gemmG_threads, B), B, 0, stream>>>(pooled, Wf1, bf1, hid, N_GRAPHS, 1);
  head_out<<<cdiv(N_GRAPHS * N_CLASSES, B), B, 0, stream>>>(hid, Wf2, bf2, out, N_GRAPHS);
}